// CompositeEmbeddingA_7413113553177
// MI455X (gfx1250) — compile-verified
//
#include <hip/hip_runtime.h>

typedef float v8f __attribute__((ext_vector_type(8)));
typedef float v2f __attribute__((ext_vector_type(2)));

static constexpr int S_TOT   = 120136;  // tokens per batch item (input)
static constexpr int OUT_TOK = 25672;   // tokens per batch item (output)
static constexpr int NB      = 4;       // batch

// ---------------------------------------------------------------------------
// on-the-fly embedding: 8 consecutive channels [e0, e0+8) of token `tok`
// emb = val_tab[value] + dep_tab[depth] + pos_tab[0][p0] + pos_tab[1][p1] + pos_tab[2][p2]
// ---------------------------------------------------------------------------
__device__ __forceinline__ void emb8(
    const int* __restrict__ value, const int* __restrict__ depth,
    const int* __restrict__ position,
    const float* __restrict__ val_tab, const float* __restrict__ dep_tab,
    const float* __restrict__ pos_tab,
    size_t tok, int e0, float4& o0, float4& o1)
{
    int v  = value[tok];
    int dp = depth[tok];
    int p0 = position[tok * 3 + 0];
    int p1 = position[tok * 3 + 1];
    int p2 = position[tok * 3 + 2];
    const float4* t0 = (const float4*)(val_tab + (size_t)v * 256 + e0);
    const float4* t1 = (const float4*)(dep_tab + (size_t)dp * 256 + e0);
    const float4* t2 = (const float4*)(pos_tab + ((size_t)p0) * 256 + e0);
    const float4* t3 = (const float4*)(pos_tab + ((size_t)257 + p1) * 256 + e0);
    const float4* t4 = (const float4*)(pos_tab + ((size_t)514 + p2) * 256 + e0);
    float4 a = t0[0], b = t1[0], c = t2[0], d = t3[0], f = t4[0];
    o0 = make_float4(a.x + b.x + c.x + d.x + f.x,
                     a.y + b.y + c.y + d.y + f.y,
                     a.z + b.z + c.z + d.z + f.z,
                     a.w + b.w + c.w + d.w + f.w);
    a = t0[1]; b = t1[1]; c = t2[1]; d = t3[1]; f = t4[1];
    o1 = make_float4(a.x + b.x + c.x + d.x + f.x,
                     a.y + b.y + c.y + d.y + f.y,
                     a.z + b.z + c.z + d.z + f.z,
                     a.w + b.w + c.w + d.w + f.w);
}

// ---------------------------------------------------------------------------
// prefix scan of (value==2) over one segment of one batch item
// idx[i] = max(cumsum_inclusive(mask)[i] - 1, 0)
// ---------------------------------------------------------------------------
__global__ __launch_bounds__(1024) void scan_mask_kernel(
    const int* __restrict__ value, int* __restrict__ idx, int segOff, int L)
{
    __shared__ int part[1024];
    const int b   = blockIdx.x;
    const int tid = threadIdx.x;
    const int chunk = L / 1024;
    const int* v = value + (size_t)b * S_TOT + segOff;
    const int base = tid * chunk;

    int cnt = 0;
    for (int i = 0; i < chunk; ++i) cnt += (v[base + i] == 2);
    part[tid] = cnt;
    __syncthreads();
    for (int off = 1; off < 1024; off <<= 1) {
        int add = (tid >= off) ? part[tid - off] : 0;
        __syncthreads();
        part[tid] += add;
        __syncthreads();
    }
    int run = part[tid] - cnt;  // exclusive prefix
    int* o = idx + (size_t)b * L;
    for (int i = 0; i < chunk; ++i) {
        run += (v[base + i] == 2);
        int ix = run - 1;
        o[base + i] = ix < 0 ? 0 : ix;
    }
}

// ---------------------------------------------------------------------------
// direct embedding for depths 1..3 -> d_out tokens [0, 328)
// ---------------------------------------------------------------------------
__global__ __launch_bounds__(256) void embed_direct_kernel(
    const int* __restrict__ value, const int* __restrict__ depth,
    const int* __restrict__ position,
    const float* __restrict__ val_tab, const float* __restrict__ dep_tab,
    const float* __restrict__ pos_tab,
    float* __restrict__ out, int nTok)
{
    const int tid  = threadIdx.x;
    const int tkn  = blockIdx.x * 8 + (tid >> 5);
    const int lane = tid & 31;
    const int e0   = lane * 8;
    const int b = tkn / nTok;
    const int t = tkn - b * nTok;
    float4 o0, o1;
    emb8(value, depth, position, val_tab, dep_tab, pos_tab,
         (size_t)b * S_TOT + t, e0, o0, o1);
    float* dst = out + ((size_t)b * OUT_TOK + t) * 256 + e0;
    ((float4*)dst)[0] = o0;
    ((float4*)dst)[1] = o1;
}

// ---------------------------------------------------------------------------
// substitution output: out[t] = (value==2) ? child[idx[t]] : emb(base token t)
// ---------------------------------------------------------------------------
__global__ __launch_bounds__(256) void sub_out_kernel(
    const int* __restrict__ value, const int* __restrict__ depth,
    const int* __restrict__ position,
    const float* __restrict__ val_tab, const float* __restrict__ dep_tab,
    const float* __restrict__ pos_tab,
    const int* __restrict__ idxArr, const float* __restrict__ child,
    float* __restrict__ out, int segOff, int L, int Gc, int outOff)
{
    const int tid  = threadIdx.x;
    const int g    = blockIdx.x * 8 + (tid >> 5);
    const int lane = tid & 31;
    const int e0   = lane * 8;
    const int b = g / L;
    const int t = g - b * L;
    const size_t tok = (size_t)b * S_TOT + segOff + t;

    float4 o0, o1;
    if (value[tok] == 2) {
        int ix = idxArr[(size_t)b * L + t];
        if (ix > Gc - 1) ix = Gc - 1;
        const float4* src = (const float4*)(child + ((size_t)(b * Gc + ix)) * 256 + e0);
        o0 = src[0];
        o1 = src[1];
    } else {
        emb8(value, depth, position, val_tab, dep_tab, pos_tab, tok, e0, o0, o1);
    }
    float* dst = out + ((size_t)b * OUT_TOK + outOff + t) * 256 + e0;
    ((float4*)dst)[0] = o0;
    ((float4*)dst)[1] = o1;
}

// ---------------------------------------------------------------------------
// WMMA f32 GEMM for the non-overlapping conv:
//   out[r, o] = sum_j A[r, j] * W[j, o] + bias[o],  r = b*G + g, j = kk*256 + e
// A rows are *virtual*: each of the kd tokens of group g is either an
// on-the-fly embedding token or (indirect mode, value==2) a row of `child`.
// Block = 256 threads = 8 waves; tile = 32 rows x 256 cols.
// Wave w computes rows [0,32) x cols [32w, 32w+32) as 2x2 WMMA 16x16 tiles.
// ---------------------------------------------------------------------------
__global__ __launch_bounds__(256) void conv_gemm_kernel(
    const int* __restrict__ value, const int* __restrict__ depth,
    const int* __restrict__ position,
    const float* __restrict__ val_tab, const float* __restrict__ dep_tab,
    const float* __restrict__ pos_tab,
    const float* __restrict__ W, const float* __restrict__ bias,
    const int* __restrict__ idxArr, const float* __restrict__ child,
    float* __restrict__ out,
    int kd, int segOff, int G, int Gc, int outStride, int outOff, int indirect)
{
    __shared__ float Atile[32 * 68];    // 32 rows x 64 K, stride 68 (conflict-free)

    const int tid     = threadIdx.x;
    const int rowBase = blockIdx.x * 32;
    const int K = kd * 256;
    const int L = G * kd;

    const int lane    = tid & 31;
    const int w       = tid >> 5;
    const int laneLow = lane & 15;
    const int hi      = lane >> 4;       // 0: K{0,1} / M{0..7}, 1: K{2,3} / M{8..15}
    const int colBase = w * 32;

    const int fRow = tid >> 3;           // 0..31  (A-fill row)
    const int fCol = (tid & 7) * 8;      // 0..56  (A-fill col, 8 floats each)

    v8f acc[2][2] = {};

    for (int kc = 0; kc < K; kc += 64) {
        // ---- stage A tile into LDS (embedding or child-row gather) ----
        {
            const int r  = rowBase + fRow;
            const int b  = r / G;
            const int g  = r - b * G;
            const int kt = kc >> 8;             // token slot within group
            const int t  = g * kd + kt;
            const int e0 = (kc & 255) + fCol;   // channel within token
            const size_t tok = (size_t)b * S_TOT + segOff + t;

            float4 o0, o1;
            bool from_child = false;
            if (indirect && value[tok] == 2) {
                int ix = idxArr[(size_t)b * L + t];
                if (ix > Gc - 1) ix = Gc - 1;
                const float4* src =
                    (const float4*)(child + ((size_t)(b * Gc + ix)) * 256 + e0);
                o0 = src[0];
                o1 = src[1];
                from_child = true;
            }
            if (!from_child)
                emb8(value, depth, position, val_tab, dep_tab, pos_tab, tok, e0, o0, o1);

            float* dst = &Atile[fRow * 68 + fCol];
            ((float4*)dst)[0] = o0;
            ((float4*)dst)[1] = o1;
        }
        __syncthreads();

        // ---- 16 K-steps of 4, 4 WMMAs each ----
#pragma unroll
        for (int kk = 0; kk < 16; ++kk) {
            const int k = kk * 4;
            // A frag (16x4 f32 layout): lanes 0-15 -> K{k,k+1}, lanes 16-31 -> K{k+2,k+3}
            v2f a0 = *(const v2f*)&Atile[laneLow * 68 + k + hi * 2];
            v2f a1 = *(const v2f*)&Atile[(16 + laneLow) * 68 + k + hi * 2];
            // B frag (4x16 f32): lanes 0-15 rows {k,k+1}, lanes 16-31 rows {k+2,k+3}
            const float* Wp = W + (size_t)(kc + k + hi * 2) * 256 + colBase + laneLow;
            v2f b0, b1;
            b0.x = Wp[0];       b0.y = Wp[256];
            b1.x = Wp[16];      b1.y = Wp[16 + 256];

            acc[0][0] = __builtin_amdgcn_wmma_f32_16x16x4_f32(false, a0, false, b0,
                                                              (short)0, acc[0][0], false, false);
            acc[0][1] = __builtin_amdgcn_wmma_f32_16x16x4_f32(false, a0, false, b1,
                                                              (short)0, acc[0][1], false, false);
            acc[1][0] = __builtin_amdgcn_wmma_f32_16x16x4_f32(false, a1, false, b0,
                                                              (short)0, acc[1][0], false, false);
            acc[1][1] = __builtin_amdgcn_wmma_f32_16x16x4_f32(false, a1, false, b1,
                                                              (short)0, acc[1][1], false, false);
        }
        __syncthreads();
    }

    // ---- bias + store (C layout: vgpr v = row v (lanes 0-15) / row v+8 (16-31)) ----
#pragma unroll
    for (int m = 0; m < 2; ++m) {
#pragma unroll
        for (int n = 0; n < 2; ++n) {
            const int col = colBase + n * 16 + laneLow;
            const float bv = bias[col];
#pragma unroll
            for (int v = 0; v < 8; ++v) {
                const int r = rowBase + m * 16 + hi * 8 + v;
                const int b = r / G;
                const int g = r - b * G;
                const size_t orow = (size_t)b * outStride + outOff + g;
                out[orow * 256 + col] = acc[m][n][v] + bv;
            }
        }
    }
}

// ---------------------------------------------------------------------------
// host-side launch
// ---------------------------------------------------------------------------
extern "C" void kernel_launch(void* const* d_in, const int* in_sizes, int n_in,
                              void* d_out, int out_size, void* d_ws, size_t ws_size,
                              hipStream_t stream)
{
    (void)in_sizes; (void)n_in; (void)out_size; (void)ws_size;

    const int*   value    = (const int*)d_in[0];
    const int*   depth    = (const int*)d_in[1];
    const int*   position = (const int*)d_in[2];
    const float* val_tab  = (const float*)d_in[3];
    const float* dep_tab  = (const float*)d_in[4];
    const float* pos_tab  = (const float*)d_in[5];
    const float* W4  = (const float*)d_in[6];   const float* b4  = (const float*)d_in[7];
    const float* W5  = (const float*)d_in[8];   const float* b5  = (const float*)d_in[9];
    const float* W6  = (const float*)d_in[10];  const float* b6  = (const float*)d_in[11];
    const float* W7a = (const float*)d_in[12];  const float* b7a = (const float*)d_in[13];
    const float* W7b = (const float*)d_in[14];  const float* b7b = (const float*)d_in[15];
    const float* W8a = (const float*)d_in[16];  const float* b8a = (const float*)d_in[17];
    const float* W8b = (const float*)d_in[18];  const float* b8b = (const float*)d_in[19];
    float* out = (float*)d_out;

    // workspace layout (~81 MB)
    char* ws = (char*)d_ws;
    size_t off = 0;
    auto alloc = [&](size_t bytes) -> void* {
        void* p = ws + off;
        off += (bytes + 255) & ~(size_t)255;
        return p;
    };
    float* C6  = (float*)alloc((size_t)NB * 2048 * 256 * 4);
    float* C7A = (float*)alloc((size_t)NB * 4096 * 256 * 4);
    float* C7B = (float*)alloc((size_t)NB * 2048 * 256 * 4);
    float* C8A = (float*)alloc((size_t)NB * 8192 * 256 * 4);
    float* C8B = (float*)alloc((size_t)NB * 4096 * 256 * 4);
    int* idx5 = (int*)alloc((size_t)NB * 4096 * 4);
    int* idx6 = (int*)alloc((size_t)NB * 16384 * 4);
    int* idx7 = (int*)alloc((size_t)NB * 32768 * 4);

    // segment token offsets: OFF = [0,8,72,328,1352,5448,21832,54600,120136]
    // output token offsets:  [0,8,72,328,584,1096,5192,9288,25672]

    // prefix scans for substitution indices
    scan_mask_kernel<<<NB, 1024, 0, stream>>>(value, idx5, 1352, 4096);
    scan_mask_kernel<<<NB, 1024, 0, stream>>>(value, idx6, 5448, 16384);
    scan_mask_kernel<<<NB, 1024, 0, stream>>>(value, idx7, 21832, 32768);

    // depths 1..3: identity embedding -> out tokens [0,328)
    embed_direct_kernel<<<(NB * 328) / 8, 256, 0, stream>>>(
        value, depth, position, val_tab, dep_tab, pos_tab, out, 328);

    // depth 4: conv(k=4) -> out tokens [328,584)
    conv_gemm_kernel<<<(NB * 256) / 32, 256, 0, stream>>>(
        value, depth, position, val_tab, dep_tab, pos_tab, W4, b4,
        nullptr, nullptr, out, 4, 328, 256, 0, OUT_TOK, 328, 0);

    // depth 5: conv(k=8) -> out tokens [584,1096)
    conv_gemm_kernel<<<(NB * 512) / 32, 256, 0, stream>>>(
        value, depth, position, val_tab, dep_tab, pos_tab, W5, b5,
        nullptr, nullptr, out, 8, 1352, 512, 0, OUT_TOK, 584, 0);

    // depth 6: conv(seg6) -> C6 ; substitute into seg5 -> out [1096,5192)
    conv_gemm_kernel<<<(NB * 2048) / 32, 256, 0, stream>>>(
        value, depth, position, val_tab, dep_tab, pos_tab, W6, b6,
        nullptr, nullptr, C6, 8, 5448, 2048, 0, 2048, 0, 0);
    sub_out_kernel<<<(NB * 4096) / 8, 256, 0, stream>>>(
        value, depth, position, val_tab, dep_tab, pos_tab,
        idx5, C6, out, 1352, 4096, 2048, 1096);

    // depth 7: conv(seg7)->C7A ; conv(s6: seg6/C7A mix)->C7B ; substitute seg5 -> out [5192,9288)
    conv_gemm_kernel<<<(NB * 4096) / 32, 256, 0, stream>>>(
        value, depth, position, val_tab, dep_tab, pos_tab, W7a, b7a,
        nullptr, nullptr, C7A, 8, 21832, 4096, 0, 4096, 0, 0);
    conv_gemm_kernel<<<(NB * 2048) / 32, 256, 0, stream>>>(
        value, depth, position, val_tab, dep_tab, pos_tab, W7b, b7b,
        idx6, C7A, C7B, 8, 5448, 2048, 4096, 2048, 0, 1);
    sub_out_kernel<<<(NB * 4096) / 8, 256, 0, stream>>>(
        value, depth, position, val_tab, dep_tab, pos_tab,
        idx5, C7B, out, 1352, 4096, 2048, 5192);

    // depth 8: conv(seg8)->C8A ; conv(s7: seg7/C8A mix)->C8B ; substitute seg6 -> out [9288,25672)
    conv_gemm_kernel<<<(NB * 8192) / 32, 256, 0, stream>>>(
        value, depth, position, val_tab, dep_tab, pos_tab, W8a, b8a,
        nullptr, nullptr, C8A, 8, 54600, 8192, 0, 8192, 0, 0);
    conv_gemm_kernel<<<(NB * 4096) / 32, 256, 0, stream>>>(
        value, depth, position, val_tab, dep_tab, pos_tab, W8b, b8b,
        idx7, C8A, C8B, 8, 21832, 4096, 8192, 4096, 0, 1);
    sub_out_kernel<<<(NB * 16384) / 8, 256, 0, stream>>>(
        value, depth, position, val_tab, dep_tab, pos_tab,
        idx6, C8B, out, 5448, 16384, 4096, 9288);
}